// OuterProductMean_15942918603426
// MI455X (gfx1250) — compile-verified
//
#include <hip/hip_runtime.h>

// Shapes fixed by reference setup
#define B_   4
#define NR_  512
#define NA_  64
#define TD_  1024
#define DD_  128
#define H_   32
#define ID_  128

typedef float v2f __attribute__((ext_vector_type(2)));
typedef float v8f __attribute__((ext_vector_type(8)));

#define LN_EPS 1e-5f

// ---------------- Phase 1a: LayerNorm(target) + W_t projection -> tf [B*NR,32]
__global__ __launch_bounds__(32) void ln_proj_t(
    const float* __restrict__ x, const float* __restrict__ g,
    const float* __restrict__ bb, const float* __restrict__ W,
    const float* __restrict__ bias, const float* __restrict__ mask,
    float* __restrict__ out) {
  __shared__ float sn[TD_];
  const int r = blockIdx.x;            // row 0..B*NR-1
  const int lane = threadIdx.x;        // 0..31
  const float* row = x + (size_t)r * TD_;
  float s = 0.f, sq = 0.f;
  #pragma unroll
  for (int j = 0; j < TD_ / 32; ++j) {
    float v = row[lane + 32 * j];
    s += v; sq += v * v;
  }
  #pragma unroll
  for (int o = 16; o > 0; o >>= 1) {
    s  += __shfl_xor(s,  o, 32);
    sq += __shfl_xor(sq, o, 32);
  }
  const float mu  = s  * (1.0f / TD_);
  const float var = sq * (1.0f / TD_) - mu * mu;
  const float rs  = rsqrtf(var + LN_EPS);
  #pragma unroll
  for (int j = 0; j < TD_ / 32; ++j) {
    int k = lane + 32 * j;
    sn[k] = (row[k] - mu) * rs * g[k] + bb[k];
  }
  __syncthreads();
  // lane == output feature h
  float acc = 0.f;
  const float* wr = W + (size_t)lane * TD_;
  #pragma unroll 4
  for (int k = 0; k < TD_; ++k) acc = fmaf(sn[k], wr[k], acc);
  out[(size_t)r * H_ + lane] = (acc + bias[lane]) * mask[r];
}

// ---------------- Phase 1b: LayerNorm(drug) + W_d projection -> df [B*NA,32]
__global__ __launch_bounds__(32) void ln_proj_d(
    const float* __restrict__ x, const float* __restrict__ g,
    const float* __restrict__ bb, const float* __restrict__ W,
    const float* __restrict__ bias, const float* __restrict__ mask,
    float* __restrict__ out) {
  __shared__ float sn[DD_];
  const int r = blockIdx.x;            // row 0..B*NA-1
  const int lane = threadIdx.x;
  const float* row = x + (size_t)r * DD_;
  float s = 0.f, sq = 0.f;
  #pragma unroll
  for (int j = 0; j < DD_ / 32; ++j) {
    float v = row[lane + 32 * j];
    s += v; sq += v * v;
  }
  #pragma unroll
  for (int o = 16; o > 0; o >>= 1) {
    s  += __shfl_xor(s,  o, 32);
    sq += __shfl_xor(sq, o, 32);
  }
  const float mu  = s  * (1.0f / DD_);
  const float var = sq * (1.0f / DD_) - mu * mu;
  const float rs  = rsqrtf(var + LN_EPS);
  #pragma unroll
  for (int j = 0; j < DD_ / 32; ++j) {
    int k = lane + 32 * j;
    sn[k] = (row[k] - mu) * rs * g[k] + bb[k];
  }
  __syncthreads();
  float acc = 0.f;
  const float* wr = W + (size_t)lane * DD_;
  #pragma unroll 4
  for (int k = 0; k < DD_; ++k) acc = fmaf(sn[k], wr[k], acc);
  out[(size_t)r * H_ + lane] = (acc + bias[lane]) * mask[r];
}

// ---------------- Phase 2: M[b][m][i][c] = sum_e df[b,m,e]*W_out[i, c*32+e]
// Stored c-contiguous (32 floats per i) so WMMA B-fragments load as float2 pairs.
__global__ __launch_bounds__(256) void build_M(
    const float* __restrict__ df, const float* __restrict__ Wout,
    float* __restrict__ M) {
  __shared__ float sdf[H_];
  const int bm = blockIdx.x;           // 0..B*NA-1
  const int t  = threadIdx.x;
  if (t < H_) sdf[t] = df[(size_t)bm * H_ + t];
  __syncthreads();
  float* mo = M + (size_t)bm * (ID_ * H_);
  #pragma unroll
  for (int j = 0; j < (ID_ * H_) / 256; ++j) {   // 16 outputs per thread
    int o = t + j * 256;                          // o = i*32 + c
    const float* wp = Wout + (size_t)o * H_;      // W_out[i, c*32 + e] contiguous in e
    float acc = 0.f;
    #pragma unroll
    for (int e = 0; e < H_; ++e) acc = fmaf(sdf[e], wp[e], acc);
    mo[o] = acc;
  }
}

// ---------------- Phase 3: out[b,n,m,i] = sum_c tf[b,n,c]*M[b,m,i,c]  (WMMA f32)
// Block = (b, 16-row n-tile, m); 8 waves, wave w covers i in [16w,16w+16).
// Per wave: D(16x16) accumulated over K=32 via 8x V_WMMA_F32_16X16X4_F32.
__global__ __launch_bounds__(256) void outer_wmma(
    const float* __restrict__ tf, const float* __restrict__ M,
    const float* __restrict__ tmask, const float* __restrict__ dmask,
    const float* __restrict__ bout, float* __restrict__ out,
    float* __restrict__ mout) {
  const int bid  = blockIdx.x;
  const int m    = bid & (NA_ - 1);          // 0..63
  const int nt   = (bid >> 6) & 31;          // 0..31
  const int b    = bid >> 11;                // 0..3
  const int lane = threadIdx.x & 31;
  const int wave = threadIdx.x >> 5;
  const int lo   = lane & 15;                // N (B,D) / M-row (A)
  const int hi   = lane >> 4;                // selects K pair {0,1} vs {2,3}
  const int n0   = nt * 16;
  const int i0   = wave * 16;

  // A frag: lane holds tf[b, n0+lo, k0 + 2*hi + {0,1}]  (contiguous pair)
  const float* ap = tf + ((size_t)(b * NR_ + n0 + lo)) * H_ + 2 * hi;
  // B frag: B[k][n] = M[b,m, i0+n, c=k]; lane holds pair at c = k0 + 2*hi
  const float* bp = M + (((size_t)(b * NA_ + m)) * ID_ + i0 + lo) * H_ + 2 * hi;

  v8f acc = {};
  #pragma unroll
  for (int k0 = 0; k0 < H_; k0 += 4) {
    v2f a  = *(const v2f*)(ap + k0);
    v2f bf = *(const v2f*)(bp + k0);
    acc = __builtin_amdgcn_wmma_f32_16x16x4_f32(
        /*neg_a=*/false, a, /*neg_b=*/false, bf,
        /*c_mod=*/(short)0, acc, /*reuse_a=*/false, /*reuse_b=*/false);
  }

  const float dm = dmask[b * NA_ + m];
  const float bo = bout[i0 + lo];
  // D layout: vgpr r, lanes 0-15 -> row r, lanes 16-31 -> row r+8; col = lo
  const size_t obase = (((size_t)(b * NR_ + n0) * NA_ + m) * ID_) + i0 + lo;
  #pragma unroll
  for (int r = 0; r < 8; ++r) {
    const int row = r + 8 * hi;
    const float tm = tmask[b * NR_ + n0 + row];
    out[obase + (size_t)row * (NA_ * ID_)] = (acc[r] + bo) * tm * dm;
  }
  // mask output (second tuple element), disjoint per block
  if (wave == 0 && lane < 16) {
    mout[(size_t)(b * NR_ + n0 + lane) * NA_ + m] = tmask[b * NR_ + n0 + lane] * dm;
  }
}

extern "C" void kernel_launch(void* const* d_in, const int* in_sizes, int n_in,
                              void* d_out, int out_size, void* d_ws, size_t ws_size,
                              hipStream_t stream) {
  const float* te     = (const float*)d_in[0];
  const float* de     = (const float*)d_in[1];
  const float* tmask  = (const float*)d_in[2];
  const float* dmask  = (const float*)d_in[3];
  const float* ln_t_g = (const float*)d_in[4];
  const float* ln_t_b = (const float*)d_in[5];
  const float* ln_d_g = (const float*)d_in[6];
  const float* ln_d_b = (const float*)d_in[7];
  const float* W_t    = (const float*)d_in[8];
  const float* b_t    = (const float*)d_in[9];
  const float* W_d    = (const float*)d_in[10];
  const float* b_d    = (const float*)d_in[11];
  const float* W_out  = (const float*)d_in[12];
  const float* b_out  = (const float*)d_in[13];

  float* out = (float*)d_out;
  float* ws  = (float*)d_ws;
  float* tf  = ws;                                   // B*NR*32   = 65536 floats
  float* df  = ws + (size_t)B_ * NR_ * H_;           // B*NA*32   = 8192 floats
  float* M   = df + (size_t)B_ * NA_ * H_;           // B*NA*128*32 = 1,048,576 floats
  float* mout = out + (size_t)B_ * NR_ * NA_ * ID_;  // mask after outer

  ln_proj_t<<<B_ * NR_, 32, 0, stream>>>(te, ln_t_g, ln_t_b, W_t, b_t, tmask, tf);
  ln_proj_d<<<B_ * NA_, 32, 0, stream>>>(de, ln_d_g, ln_d_b, W_d, b_d, dmask, df);
  build_M<<<B_ * NA_, 256, 0, stream>>>(df, W_out, M);
  outer_wmma<<<B_ * (NR_ / 16) * NA_, 256, 0, stream>>>(tf, M, tmask, dmask, b_out,
                                                        out, mout);
}